// net_gcn_multitask_85864986181826
// MI455X (gfx1250) — compile-verified
//
#include <hip/hip_runtime.h>
#include <stdint.h>

// ---------------- problem constants ----------------
#define N_NODES    100000
#define N_EDGES    1600000
#define DIM        128
#define MASK_TOTAL (N_NODES * DIM)      // 12,800,000 mask elements
#define MASK_HALF  (MASK_TOTAL / 2)     // 6,400,000
#define KEEP       0.8f
#define INV_KEEP   1.25f
#define MTILES     (N_NODES / 16)       // 6250 (exact)
#define NC_H       40                    // W1 out cols
#define NC_S       64                    // Wss out cols
#define LDS_STRIDE 136                   // padded row stride (floats), keeps b64 align
#define LDS_PANEL  (16 * LDS_STRIDE)

typedef float v2f __attribute__((ext_vector_type(2)));
typedef float v8f __attribute__((ext_vector_type(8)));

// ---------------- threefry2x32-20 (JAX PRNG) ----------------
__host__ __device__ inline uint32_t rotl32(uint32_t x, int r) {
  return (x << r) | (x >> (32 - r));
}

__host__ __device__ inline void threefry2x32(uint32_t k0, uint32_t k1,
                                             uint32_t c0, uint32_t c1,
                                             uint32_t* o0, uint32_t* o1) {
  uint32_t ks2 = 0x1BD11BDAu ^ k0 ^ k1;
  uint32_t x0 = c0 + k0, x1 = c1 + k1;
#define TF_R(r) { x0 += x1; x1 = rotl32(x1, (r)); x1 ^= x0; }
  TF_R(13) TF_R(15) TF_R(26) TF_R(6)   x0 += k1;  x1 += ks2 + 1u;
  TF_R(17) TF_R(29) TF_R(16) TF_R(24)  x0 += ks2; x1 += k0  + 2u;
  TF_R(13) TF_R(15) TF_R(26) TF_R(6)   x0 += k0;  x1 += k1  + 3u;
  TF_R(17) TF_R(29) TF_R(16) TF_R(24)  x0 += k1;  x1 += ks2 + 4u;
  TF_R(13) TF_R(15) TF_R(26) TF_R(6)   x0 += ks2; x1 += k0  + 5u;
#undef TF_R
  *o0 = x0; *o1 = x1;
}

// bernoulli(keep) at flat index j of the (N,128) mask: JAX iota-halving
// counter layout + (bits>>9 | 0x3f800000) - 1 uniform construction.
__device__ inline bool keep_mask(uint32_t k0, uint32_t k1, uint32_t j) {
  uint32_t o0, o1, w;
  if (j < (uint32_t)MASK_HALF) { threefry2x32(k0, k1, j, j + MASK_HALF, &o0, &o1); w = o0; }
  else                         { threefry2x32(k0, k1, j - MASK_HALF, j, &o0, &o1); w = o1; }
  float u = __uint_as_float(0x3f800000u | (w >> 9)) - 1.0f;
  return u < KEEP;
}

// ---------------- zero fill ----------------
__global__ void zero_f4(float4* __restrict__ p, int n4) {
  int i = blockIdx.x * blockDim.x + threadIdx.x;
  if (i < n4) p[i] = float4{0.0f, 0.0f, 0.0f, 0.0f};
}

// ---------------- stage 1 SpMM: A = spmm(x), one wave per edge ----------------
__global__ void spmm_edges1(const int* __restrict__ er, const int* __restrict__ ec,
                            const float* __restrict__ ev,
                            const float* __restrict__ src, float* __restrict__ dst) {
  int w    = (blockIdx.x * blockDim.x + threadIdx.x) >> 5;
  int lane = threadIdx.x & 31;
  if (w >= N_EDGES) return;
  int   r = er[w], c = ec[w];
  float v = ev[w];
  float4 d = ((const float4*)(src + (size_t)c * DIM))[lane];   // b128 gather (L2-resident)
  float* o = dst + (size_t)r * DIM + lane * 4;
  __hip_atomic_fetch_add(o + 0, v * d.x, __ATOMIC_RELAXED, __HIP_MEMORY_SCOPE_AGENT);
  __hip_atomic_fetch_add(o + 1, v * d.y, __ATOMIC_RELAXED, __HIP_MEMORY_SCOPE_AGENT);
  __hip_atomic_fetch_add(o + 2, v * d.z, __ATOMIC_RELAXED, __HIP_MEMORY_SCOPE_AGENT);
  __hip_atomic_fetch_add(o + 3, v * d.w, __ATOMIC_RELAXED, __HIP_MEMORY_SCOPE_AGENT);
}

// ---------------- stage 2 (fused): Z=relu(A@W0) -> dropout -> LDS -> P,Q ----------------
// block = 256 threads = 8 waves = one 16-row panel.
// phase A: wave wv computes the 16x16 Z tile at cols wv*16 (32x V_WMMA_F32_16X16X4_F32),
//          applies relu + both dropout masks, stores h1/s1 tiles to LDS.
// phase B: waves 0..2 compute P = h1@W1 (40 cols), waves 3..6 compute Q = s1@Wss (64 cols),
//          reading A-fragments from LDS (ds path).
__global__ void mid_fused(const float* __restrict__ A,  const float* __restrict__ W0,
                          const float* __restrict__ W1, const float* __restrict__ Wss,
                          float* __restrict__ P, float* __restrict__ Q,
                          uint32_t k1a, uint32_t k1b, uint32_t k2a, uint32_t k2b) {
  __shared__ float lds[2 * LDS_PANEL];   // [0]=h1 panel, [1]=s1 panel, padded stride

  int wv   = threadIdx.x >> 5;           // 0..7
  int mt   = blockIdx.x;                 // 0..6249
  int lane = threadIdx.x & 31;
  int g    = lane >> 4;                  // half-wave group
  int ln   = lane & 15;
  int row0 = mt << 4;

  // ---- phase A ----
  {
    int col0 = wv << 4;
    // A frag: lane holds A[row0+ln][k+2g], A[row0+ln][k+2g+1]
    const float* arow = A  + (size_t)(row0 + ln) * DIM + (g << 1);
    // B frag: lane holds W0[k+2g][col0+ln], W0[k+2g+1][col0+ln]
    const float* bcol = W0 + (size_t)(g << 1) * DIM + col0 + ln;
    v8f acc = {};
#pragma unroll
    for (int k = 0; k < DIM; k += 4) {
      v2f a; a.x = arow[k];                 a.y = arow[k + 1];
      v2f b; b.x = bcol[(size_t)k * DIM];   b.y = bcol[(size_t)k * DIM + DIM];
      acc = __builtin_amdgcn_wmma_f32_16x16x4_f32(false, a, false, b,
                                                  (short)0, acc, false, false);
    }
#pragma unroll
    for (int r = 0; r < 8; ++r) {
      int prow = r + (g << 3);                       // 0..15 panel row
      int col  = col0 + ln;                          // 0..127
      float z  = acc[r] > 0.0f ? acc[r] * INV_KEEP : 0.0f;   // relu + /keep
      uint32_t j = (uint32_t)(row0 + prow) * DIM + (uint32_t)col;
      lds[            prow * LDS_STRIDE + col] = keep_mask(k1a, k1b, j) ? z : 0.0f;
      lds[LDS_PANEL + prow * LDS_STRIDE + col] = keep_mask(k2a, k2b, j) ? z : 0.0f;
    }
  }
  __syncthreads();

  // ---- phase B: 7 head tiles (3x W1 + 4x Wss) on waves 0..6 ----
  if (wv < 7) {
    const float* W; float* o; int ncols, col0, soff;
    if (wv < 3) { W = W1;  o = P; ncols = NC_H; col0 = wv << 4;        soff = 0; }
    else        { W = Wss; o = Q; ncols = NC_S; col0 = (wv - 3) << 4;  soff = LDS_PANEL; }

    int bc = col0 + ln; if (bc >= ncols) bc = ncols - 1;   // clamp; lanes masked at store
    int abase = soff + ln * LDS_STRIDE + (g << 1);         // LDS A-fragment base
    const float* bcol = W + (size_t)(g << 1) * ncols + bc;

    v8f acc = {};
#pragma unroll
    for (int k = 0; k < DIM; k += 4) {
      v2f a; a.x = lds[abase + k];            a.y = lds[abase + k + 1];   // ds_load
      v2f b; b.x = bcol[(size_t)k * ncols];   b.y = bcol[(size_t)k * ncols + ncols];
      acc = __builtin_amdgcn_wmma_f32_16x16x4_f32(false, a, false, b,
                                                  (short)0, acc, false, false);
    }
    int col = col0 + ln;
    if (col < ncols) {
#pragma unroll
      for (int r = 0; r < 8; ++r) {
        int row = row0 + r + (g << 3);
        o[(size_t)row * ncols + col] = acc[r];
      }
    }
  }
}

// ---------------- stage 3 SpMM: h = spmm(P) [40], s = spmm(Q) [64] -> d_out ----------------
// one wave per edge; float2 per lane (Q: 32x2=64 exact; P: lanes 0..19 cover 40)
__global__ void spmm_heads(const int* __restrict__ er, const int* __restrict__ ec,
                           const float* __restrict__ ev,
                           const float* __restrict__ P, const float* __restrict__ Q,
                           float* __restrict__ outh, float* __restrict__ outs) {
  int w    = (blockIdx.x * blockDim.x + threadIdx.x) >> 5;
  int lane = threadIdx.x & 31;
  if (w >= N_EDGES) return;
  int   r = er[w], c = ec[w];
  float v = ev[w];

  float2 q = ((const float2*)(Q + (size_t)c * NC_S))[lane];
  float* oq = outs + (size_t)r * NC_S + lane * 2;
  __hip_atomic_fetch_add(oq + 0, v * q.x, __ATOMIC_RELAXED, __HIP_MEMORY_SCOPE_AGENT);
  __hip_atomic_fetch_add(oq + 1, v * q.y, __ATOMIC_RELAXED, __HIP_MEMORY_SCOPE_AGENT);

  if (lane < NC_H / 2) {
    float2 p = ((const float2*)(P + (size_t)c * NC_H))[lane];
    float* op = outh + (size_t)r * NC_H + lane * 2;
    __hip_atomic_fetch_add(op + 0, v * p.x, __ATOMIC_RELAXED, __HIP_MEMORY_SCOPE_AGENT);
    __hip_atomic_fetch_add(op + 1, v * p.y, __ATOMIC_RELAXED, __HIP_MEMORY_SCOPE_AGENT);
  }
}

// ---------------- launch ----------------
extern "C" void kernel_launch(void* const* d_in, const int* in_sizes, int n_in,
                              void* d_out, int out_size, void* d_ws, size_t ws_size,
                              hipStream_t stream) {
  const float* x   = (const float*)d_in[0];
  const int*   er  = (const int*)  d_in[1];
  const int*   ec  = (const int*)  d_in[2];
  const float* ev  = (const float*)d_in[3];
  const float* W0  = (const float*)d_in[4];
  const float* W1  = (const float*)d_in[5];
  const float* Wss = (const float*)d_in[6];
  float* out = (float*)d_out;

  // workspace: A [N,128], P [N,40], Q [N,64]  (92.8 MB total)
  float* A = (float*)d_ws;
  float* P = A + (size_t)MASK_TOTAL;
  float* Q = P + (size_t)N_NODES * NC_H;

  // host-side: jax.random.split(jax.random.key(42)) -> k1 (m1), k2 (m2)
  uint32_t w0a, w1a, w0b, w1b;
  threefry2x32(0u, 42u, 0u, 2u, &w0a, &w1a);
  threefry2x32(0u, 42u, 1u, 3u, &w0b, &w1b);
  uint32_t k1a = w0a, k1b = w0b;   // first split key  (mask m1)
  uint32_t k2a = w1a, k2b = w1b;   // second split key (mask m2)

  const int nA4         = MASK_TOTAL / 4;                       // 3.2M float4
  const int zeroA_blk   = (nA4 + 255) / 256;
  const int nO4         = (N_NODES * (NC_H + NC_S)) / 4;        // 2.6M float4
  const int zeroO_blk   = (nO4 + 255) / 256;
  const int spmm_blocks = (N_EDGES * 32 + 255) / 256;           // one wave per edge

  // stage 1: A = spmm(x)
  zero_f4<<<zeroA_blk, 256, 0, stream>>>((float4*)A, nA4);
  spmm_edges1<<<spmm_blocks, 256, 0, stream>>>(er, ec, ev, x, A);

  // stage 2 (fused): Z = relu(A@W0); dropout in LDS; P = h1@W1; Q = s1@Wss
  mid_fused<<<MTILES, 256, 0, stream>>>(A, W0, W1, Wss, P, Q, k1a, k1b, k2a, k2b);

  // stage 3: h = spmm(P), s = spmm(Q) scattered directly into d_out
  zero_f4<<<zeroO_blk, 256, 0, stream>>>((float4*)out, nO4);
  spmm_heads<<<spmm_blocks, 256, 0, stream>>>(er, ec, ev, P, Q,
                                              out, out + (size_t)N_NODES * NC_H);
}